// MultitaskLoss_16836271800936
// MI455X (gfx1250) — compile-verified
//
#include <hip/hip_runtime.h>
#include <math.h>

typedef __attribute__((ext_vector_type(2))) float v2f;
typedef __attribute__((ext_vector_type(8))) float v8f;

#define BN 4096
#define DK 256
#define TILE 16
#define JSTEP 32             // stage 32 j-rows per iteration, 2 accumulators/wave
#define NJT (BN / JSTEP)     // 128
#define NKSTEP (DK / 4)      // 64 WMMA f32 16x16x4 steps per 16x16 tile
#define LDSB_STRIDE 260      // 256 + 4 pad -> conflict-free fragment reads
#define SIM_MAX 10.0f        // |cos|/T <= 1/0.1 : exact bound for normalized rows

// workspace layout (floats)
#define ACC_CONT 0
#define ACC_CLS  1
#define ACC_S1   2
#define ACC_S0   3
#define ACC_CE   4
#define ACC_PD   5
#define ACC_CT   6
#define LSE_OFF  64
#define PN_OFF   (64 + BN)   // normalized (and 1/sqrt(T)-scaled) projections

__global__ void init_kernel(float* ws) {
    if (threadIdx.x < 8) ws[threadIdx.x] = 0.0f;
}

// Normalize rows of projections and fold in 1/sqrt(temp): sim = (p_i/|p_i|)·(p_j/|p_j|)/T
__global__ __launch_bounds__(256) void norm_kernel(const float* __restrict__ proj,
                                                   float* __restrict__ ws) {
    const int wave = threadIdx.x >> 5;
    const int lane = threadIdx.x & 31;
    const int row  = blockIdx.x * 8 + wave;
    const float* src = proj + row * DK;
    float* dst = ws + PN_OFF + row * DK;

    float vals[8];
    float ss = 0.0f;
#pragma unroll
    for (int i = 0; i < 8; ++i) {
        float v = src[lane + i * 32];
        vals[i] = v;
        ss += v * v;
    }
#pragma unroll
    for (int m = 16; m; m >>= 1) ss += __shfl_xor(ss, m, 32);
    float nrm = fmaxf(sqrtf(ss), 1e-12f);
    float inv = (1.0f / nrm) * 3.16227766016837933f;   // 1/sqrt(0.1)
#pragma unroll
    for (int i = 0; i < 8; ++i) dst[lane + i * 32] = vals[i] * inv;
}

// A-fragment (16x4 f32): lane&15 -> M row, lane>>4 selects K pair {0,1} vs {2,3}
__device__ inline v2f load_fragG(const float* __restrict__ base, int row0, int lane, int ks) {
    const float* p = base + (row0 + (lane & 15)) * DK + ks * 4 + ((lane >> 4) << 1);
    return *(const v2f*)p;
}
__device__ inline v2f load_fragL(const float* lds, int lane, int ks) {
    const float* p = lds + (lane & 15) * LDSB_STRIDE + ks * 4 + ((lane >> 4) << 1);
    return *(const v2f*)p;
}

// Cooperative stage of 32 rows x 256 cols of p_norm into LDS (+ prefetch of next block)
__device__ inline void stage_jblock(const float* __restrict__ pn, float* ldsB, int j0) {
    const int r  = threadIdx.x >> 3;          // 0..31
    const int c0 = (threadIdx.x & 7) * 32;    // 0..224
    const float4* g = (const float4*)(pn + (j0 + r) * DK + c0);
    float4* s4 = (float4*)(ldsB + r * LDSB_STRIDE + c0);
#pragma unroll
    for (int q = 0; q < 8; ++q) s4[q] = g[q];
    if (j0 + JSTEP < BN)
        __builtin_prefetch(pn + (j0 + JSTEP + r) * DK + c0, 0, 1);
}

// Pass A: lse over negatives with FIXED max (sim bounded by +/-10), diag -> exp==0
__global__ __launch_bounds__(256) void lse_kernel(const int* __restrict__ labels,
                                                  float* __restrict__ ws) {
    __shared__ float ldsB[JSTEP * LDSB_STRIDE];
    const float* pn = ws + PN_OFF;
    float* lse = ws + LSE_OFF;
    const int wave = threadIdx.x >> 5;
    const int lane = threadIdx.x & 31;
    const int i0 = (blockIdx.x * 8 + wave) * TILE;
    const int hi = lane >> 4;
    const int nlo = lane & 15;

    int labi[8];
    float esum[8];
#pragma unroll
    for (int v = 0; v < 8; ++v) { labi[v] = labels[i0 + v + 8 * hi]; esum[v] = 0.0f; }

    for (int jt = 0; jt < NJT; ++jt) {
        const int j0 = jt * JSTEP;
        stage_jblock(pn, ldsB, j0);
        __syncthreads();

        v8f acc0 = {}, acc1 = {};
#pragma unroll 8
        for (int ks = 0; ks < NKSTEP; ++ks) {
            v2f a  = load_fragG(pn, i0, lane, ks);
            v2f b0 = load_fragL(ldsB, lane, ks);
            v2f b1 = load_fragL(ldsB + TILE * LDSB_STRIDE, lane, ks);
            acc0 = __builtin_amdgcn_wmma_f32_16x16x4_f32(false, a, false, b0,
                                                         (short)0, acc0, false, false);
            acc1 = __builtin_amdgcn_wmma_f32_16x16x4_f32(false, a, false, b1,
                                                         (short)0, acc1, false, false);
        }

        const int labA = labels[j0 + nlo];
        const int labB = labels[j0 + TILE + nlo];
#pragma unroll
        for (int v = 0; v < 8; ++v) {
            // label mismatch -> negative pair -> include; equal labels (incl. diag) -> 0
            float e0 = (labi[v] != labA) ? __expf(acc0[v] - SIM_MAX) : 0.0f;
            float e1 = (labi[v] != labB) ? __expf(acc1[v] - SIM_MAX) : 0.0f;
            esum[v] += e0 + e1;
        }
        __syncthreads();
    }
    // one cross-lane (16-wide) reduction per row at the very end
#pragma unroll
    for (int v = 0; v < 8; ++v) {
        float s = esum[v];
#pragma unroll
        for (int m = 8; m; m >>= 1) s += __shfl_xor(s, m, 16);
        if (nlo == 0) lse[i0 + v + 8 * hi] = SIM_MAX + __logf(s);
    }
}

// Pass B: sum over positive pairs of logaddexp(sim, lse_i) - sim = log1p(exp(lse_i - sim))
__global__ __launch_bounds__(256) void pair_kernel(const int* __restrict__ labels,
                                                   float* __restrict__ ws) {
    __shared__ float ldsB[JSTEP * LDSB_STRIDE];
    const float* pn = ws + PN_OFF;
    const float* lse = ws + LSE_OFF;
    const int wave = threadIdx.x >> 5;
    const int lane = threadIdx.x & 31;
    const int i0 = (blockIdx.x * 8 + wave) * TILE;
    const int hi = lane >> 4;
    const int nlo = lane & 15;

    int labi[8];
    float lsei[8];
#pragma unroll
    for (int v = 0; v < 8; ++v) {
        const int i = i0 + v + 8 * hi;
        labi[v] = labels[i];
        lsei[v] = lse[i];
    }

    float local = 0.0f;
    for (int jt = 0; jt < NJT; ++jt) {
        const int j0 = jt * JSTEP;
        stage_jblock(pn, ldsB, j0);
        __syncthreads();

        v8f acc0 = {}, acc1 = {};
#pragma unroll 8
        for (int ks = 0; ks < NKSTEP; ++ks) {
            v2f a  = load_fragG(pn, i0, lane, ks);
            v2f b0 = load_fragL(ldsB, lane, ks);
            v2f b1 = load_fragL(ldsB + TILE * LDSB_STRIDE, lane, ks);
            acc0 = __builtin_amdgcn_wmma_f32_16x16x4_f32(false, a, false, b0,
                                                         (short)0, acc0, false, false);
            acc1 = __builtin_amdgcn_wmma_f32_16x16x4_f32(false, a, false, b1,
                                                         (short)0, acc1, false, false);
        }

        const int jA = j0 + nlo;
        const int jB = j0 + TILE + nlo;
        const int labA = labels[jA];
        const int labB = labels[jB];
#pragma unroll
        for (int v = 0; v < 8; ++v) {
            const int i = i0 + v + 8 * hi;
            if (labi[v] == labA && i != jA) {
                float d = lsei[v] - acc0[v];
                local += (d > 0.0f) ? d + log1pf(__expf(-d)) : log1pf(__expf(d));
            }
            if (labi[v] == labB && i != jB) {
                float d = lsei[v] - acc1[v];
                local += (d > 0.0f) ? d + log1pf(__expf(-d)) : log1pf(__expf(d));
            }
        }
        __syncthreads();
    }
#pragma unroll
    for (int m = 16; m; m >>= 1) local += __shfl_xor(local, m, 32);
    if (lane == 0) atomicAdd(&ws[ACC_CONT], local);
}

__device__ inline float block_sum(float v, float* smem) {
    const int lane = threadIdx.x & 31;
    const int wave = threadIdx.x >> 5;
#pragma unroll
    for (int m = 16; m; m >>= 1) v += __shfl_xor(v, m, 32);
    if (lane == 0) smem[wave] = v;
    __syncthreads();
    float r = 0.0f;
    if (wave == 0) {
        r = (lane < 8) ? smem[lane] : 0.0f;
#pragma unroll
        for (int m = 4; m; m >>= 1) r += __shfl_xor(r, m, 32);
    }
    __syncthreads();
    return r;   // valid in wave 0
}

__global__ __launch_bounds__(256) void focal_kernel(const float* __restrict__ logits,
                                                    const int* __restrict__ labels,
                                                    float* __restrict__ ws) {
    __shared__ float smem[8];
    const int b = blockIdx.x * 256 + threadIdx.x;
    const float x0 = logits[2 * b]     * (1.0f / 1.5f);
    const float x1 = logits[2 * b + 1] * (1.0f / 1.5f);
    const int lab = labels[b];
    const float m = fmaxf(x0, x1);
    const float l = m + logf(expf(x0 - m) + expf(x1 - m));
    const float ce = l - (lab ? x1 : x0);
    const float pt = expf(-ce);
    const float om = 1.0f - pt;
    float val = om * om * ce;
    val = block_sum(val, smem);
    if (threadIdx.x == 0) atomicAdd(&ws[ACC_CLS], val);
}

__global__ __launch_bounds__(256) void region_kernel(const float* __restrict__ rp,
                                                     const int* __restrict__ labels,
                                                     float* __restrict__ ws) {
    __shared__ float smem[8];
    const int b = blockIdx.x * 256 + threadIdx.x;
    const int lab = labels[b];

    float p[4][2], lpe[4][2], ent[4];
    float ce = 0.0f;
#pragma unroll
    for (int r = 0; r < 4; ++r) {
        const float x0 = rp[(r * BN + b) * 2 + 0];
        const float x1 = rp[(r * BN + b) * 2 + 1];
        const float m = fmaxf(x0, x1);
        const float l = m + logf(expf(x0 - m) + expf(x1 - m));
        const float l0 = x0 - l, l1 = x1 - l;
        const float p0 = expf(l0), p1 = expf(l1);
        p[r][0] = p0; p[r][1] = p1;
        ent[r] = (p0 > 0.0f ? p0 * logf(p0) : 0.0f) + (p1 > 0.0f ? p1 * logf(p1) : 0.0f);
        lpe[r][0] = logf(p0 + 1e-10f);
        lpe[r][1] = logf(p1 + 1e-10f);
        ce += -(lab ? l1 : l0);
    }
    float klsum = 0.0f;
#pragma unroll
    for (int i = 0; i < 4; ++i)
#pragma unroll
        for (int j = i + 1; j < 4; ++j)
            klsum += ent[j] - (p[j][0] * lpe[i][0] + p[j][1] * lpe[i][1]);

    float s1 = lab ? klsum : 0.0f;
    float s0 = lab ? 0.0f : klsum;
    float c1 = lab ? 1.0f : 0.0f;
    float c0 = lab ? 0.0f : 1.0f;

    s1 = block_sum(s1, smem);
    if (threadIdx.x == 0) atomicAdd(&ws[ACC_S1], s1);
    s0 = block_sum(s0, smem);
    if (threadIdx.x == 0) atomicAdd(&ws[ACC_S0], s0);
    ce = block_sum(ce, smem);
    if (threadIdx.x == 0) atomicAdd(&ws[ACC_CE], ce);
    c1 = block_sum(c1, smem);
    if (threadIdx.x == 0) atomicAdd(&ws[ACC_PD], c1);
    c0 = block_sum(c0, smem);
    if (threadIdx.x == 0) atomicAdd(&ws[ACC_CT], c0);
}

__global__ void finalize_kernel(const float* __restrict__ ws, float* __restrict__ out) {
    if (threadIdx.x == 0 && blockIdx.x == 0) {
        const float invB = 1.0f / (float)BN;
        const float cls  = ws[ACC_CLS]  * invB;
        const float cont = ws[ACC_CONT] * invB;
        const float pd = ws[ACC_PD], ct = ws[ACC_CT];
        const float kl = (pd > 0.0f ? ws[ACC_S1] / pd : 0.0f)
                       + (ct > 0.0f ? ws[ACC_S0] / ct : 0.0f);
        const float reg = (kl + ws[ACC_CE] * invB) * 0.1f;   // /(R*(R-1)/2 + R) = /10
        out[0] = 1.0f * cls + 0.5f * cont + 0.3f * reg;
    }
}

extern "C" void kernel_launch(void* const* d_in, const int* in_sizes, int n_in,
                              void* d_out, int out_size, void* d_ws, size_t ws_size,
                              hipStream_t stream) {
    const float* logits = (const float*)d_in[0];   // [4096, 2]
    const float* proj   = (const float*)d_in[1];   // [4096, 256]
    const float* rp     = (const float*)d_in[2];   // [4, 4096, 2]
    const int*   labels = (const int*)d_in[3];     // [4096]
    float* out = (float*)d_out;
    float* ws  = (float*)d_ws;

    init_kernel<<<1, 32, 0, stream>>>(ws);
    norm_kernel<<<BN / 8, 256, 0, stream>>>(proj, ws);
    lse_kernel<<<(BN / TILE) / 8, 256, 0, stream>>>(labels, ws);
    pair_kernel<<<(BN / TILE) / 8, 256, 0, stream>>>(labels, ws);
    focal_kernel<<<BN / 256, 256, 0, stream>>>(logits, labels, ws);
    region_kernel<<<BN / 256, 256, 0, stream>>>(rp, labels, ws);
    finalize_kernel<<<1, 32, 0, stream>>>(ws, out);
}